// ArchNN_7060926234949
// MI455X (gfx1250) — compile-verified
//
#include <hip/hip_runtime.h>

typedef __attribute__((ext_vector_type(2))) float v2f;
typedef __attribute__((ext_vector_type(8))) float v8f;

#define NNODE 50000   // divisible by 16 -> no row guards in GEMM
#define NEDGE 800000

__device__ __forceinline__ float elu1(float x) {
    return x > 0.0f ? x : (__expf(x) - 1.0f);
}

// ---------------------------------------------------------------------------
// Degree / inverse degree (self-loop adds +1)
// ---------------------------------------------------------------------------
__global__ void deg_kernel(const int* __restrict__ dst, float* __restrict__ deg, int E) {
    int e = blockIdx.x * blockDim.x + threadIdx.x;
    if (e < E) atomicAdd(&deg[dst[e]], 1.0f);
}

__global__ void invdeg_kernel(float* __restrict__ deg, int n) {
    int i = blockIdx.x * blockDim.x + threadIdx.x;
    if (i < n) deg[i] = 1.0f / (deg[i] + 1.0f);   // +1 = self loop; always >= 1
}

// ---------------------------------------------------------------------------
// Zero-pad fc2 weights/bias from width 10 to width 16 so the GEMM loop is
// completely unguarded (only the epilogue store checks n < Nout).
// ---------------------------------------------------------------------------
__global__ void pad_fc2_kernel(const float* __restrict__ w, const float* __restrict__ b,
                               float* __restrict__ wpad, float* __restrict__ bpad) {
    int idx = blockIdx.x * blockDim.x + threadIdx.x;     // 256*16 elements
    if (idx < 256 * 16) {
        int k = idx >> 4, n = idx & 15;
        wpad[idx] = (n < 10) ? w[k * 10 + n] : 0.0f;
    }
    if (idx < 16) bpad[idx] = (idx < 10) ? b[idx] : 0.0f;
}

// ---------------------------------------------------------------------------
// fp32 WMMA GEMM: C[M,Nout] = (optional elu)(A[M,K] @ B[K,Nc] + bias)
// Requirements: M % 16 == 0, K % 4 == 0, Nc % (16*TN) == 0, Nout <= Nc.
// One wave computes a 16 x (16*TN) strip of C: one A fragment (float2, b64
// load) feeds TN v_wmma_f32_16x16x4_f32 per K-step. No guards in the loop.
// ---------------------------------------------------------------------------
template <int TN, bool ELU>
__global__ void gemm_wmma(const float* __restrict__ A, const float* __restrict__ B,
                          const float* __restrict__ bias, float* __restrict__ C,
                          int M, int K, int Nc, int Nout) {
    const int lane = threadIdx.x & 31;
    const int wave = (blockIdx.x * blockDim.x + threadIdx.x) >> 5;
    const int groupsN = Nc / (16 * TN);
    const int tilesM  = M >> 4;
    if (wave >= tilesM * groupsN) return;     // wave-uniform: EXEC stays all-1s

    const int rowTile = wave / groupsN;
    const int colGrp  = wave % groupsN;
    const int l16 = lane & 15;
    const int hi  = lane >> 4;                // 0 or 1 (selects K pair)
    const int r = rowTile * 16 + l16;         // A row handled by this lane

    const float2* __restrict__ A2 = (const float2*)(A + r * K);   // K even
    int ncol[TN];
#pragma unroll
    for (int t = 0; t < TN; ++t) ncol[t] = colGrp * 16 * TN + t * 16 + l16;

    v8f acc[TN];
#pragma unroll
    for (int t = 0; t < TN; ++t) acc[t] = (v8f){};

    for (int k0 = 0; k0 < K; k0 += 4) {
        const int ka = k0 + hi * 2;
        float2 af = A2[(k0 >> 1) + hi];       // A[r][ka], A[r][ka+1] : one b64
        v2f a; a.x = af.x; a.y = af.y;
        const float* __restrict__ Brow0 = B + ka * Nc;
        const float* __restrict__ Brow1 = Brow0 + Nc;
#pragma unroll
        for (int t = 0; t < TN; ++t) {
            v2f b; b.x = Brow0[ncol[t]]; b.y = Brow1[ncol[t]];
            acc[t] = __builtin_amdgcn_wmma_f32_16x16x4_f32(
                false, a, false, b, (short)0, acc[t], false, false);
        }
    }

#pragma unroll
    for (int t = 0; t < TN; ++t) {
        const int n = ncol[t];
        const float bv = (bias != nullptr) ? bias[n] : 0.0f;
#pragma unroll
        for (int v = 0; v < 8; ++v) {
            const int row = rowTile * 16 + v + hi * 8;   // C frag: VGPR v, half
            if (n < Nout) {
                float x = acc[t][v] + bv;
                C[row * Nout + n] = ELU ? elu1(x) : x;
            }
        }
    }
}

// ---------------------------------------------------------------------------
// P[i][h] = sum_k h[i,k] * U[k,h]   (H == 4, translation-invariant logits)
// ---------------------------------------------------------------------------
__global__ void proj_kernel(const float* __restrict__ h, const float* __restrict__ U,
                            float* __restrict__ P, int n, int K) {
    int i = blockIdx.x * blockDim.x + threadIdx.x;
    if (i >= n) return;
    const float* hr = h + i * K;
    float p0 = 0.f, p1 = 0.f, p2 = 0.f, p3 = 0.f;
    for (int k = 0; k < K; ++k) {
        float x = hr[k];
        p0 += x * U[k * 4 + 0];
        p1 += x * U[k * 4 + 1];
        p2 += x * U[k * 4 + 2];
        p3 += x * U[k * 4 + 3];
    }
    P[i * 4 + 0] = p0; P[i * 4 + 1] = p1; P[i * 4 + 2] = p2; P[i * 4 + 3] = p3;
}

// ---------------------------------------------------------------------------
// Edge scatter: each wave processes 32/(out/4) edges so every lane owns one
// float4 column slice. q = softmax(P[src]-P[dst]+c);
// agg[dst] += sum_h q_h * M[src,h,:].  128-bit gathers, fp32 atomic scatter.
// c4shift = log2(out/4)  (out in {32,64,128} -> 3,4,5)
// ---------------------------------------------------------------------------
__global__ void edge_kernel(const int* __restrict__ ei, int E, int n,
                            const float* __restrict__ P, const float* __restrict__ c,
                            const float* __restrict__ Mbuf, float* __restrict__ agg,
                            int out, int c4shift) {
    const int lane = threadIdx.x & 31;
    const int wave = blockIdx.x * (blockDim.x >> 5) + (threadIdx.x >> 5);
    const int c4   = 1 << c4shift;            // float4s per row
    const int sub  = lane >> c4shift;         // edge slot within wave
    const int i    = lane & (c4 - 1);         // float4 index within row
    const int epw  = 32 >> c4shift;           // edges per wave

    const int e = wave * epw + sub;
    if (e >= E + n) return;

    int src, dst;
    if (e < E) { src = ei[e]; dst = ei[E + e]; }
    else       { src = dst = e - E; }

    // 4-way softmax (redundant across the c4 lanes of a subgroup; cache-hot)
    float l0, l1, l2, l3;
    if (src == dst) {
        l0 = c[0]; l1 = c[1]; l2 = c[2]; l3 = c[3];
    } else {
        l0 = P[src * 4 + 0] - P[dst * 4 + 0] + c[0];
        l1 = P[src * 4 + 1] - P[dst * 4 + 1] + c[1];
        l2 = P[src * 4 + 2] - P[dst * 4 + 2] + c[2];
        l3 = P[src * 4 + 3] - P[dst * 4 + 3] + c[3];
    }
    const float mx = fmaxf(fmaxf(l0, l1), fmaxf(l2, l3));
    float q0 = __expf(l0 - mx), q1 = __expf(l1 - mx);
    float q2 = __expf(l2 - mx), q3 = __expf(l3 - mx);
    const float inv = 1.0f / (q0 + q1 + q2 + q3);
    q0 *= inv; q1 *= inv; q2 *= inv; q3 *= inv;

    const float4* __restrict__ Ms = (const float4*)(Mbuf + src * 4 * out);
    float4 m0 = Ms[i];
    float4 m1 = Ms[c4 + i];
    float4 m2 = Ms[2 * c4 + i];
    float4 m3 = Ms[3 * c4 + i];
    float vx = q0 * m0.x + q1 * m1.x + q2 * m2.x + q3 * m3.x;
    float vy = q0 * m0.y + q1 * m1.y + q2 * m2.y + q3 * m3.y;
    float vz = q0 * m0.z + q1 * m1.z + q2 * m2.z + q3 * m3.z;
    float vw = q0 * m0.w + q1 * m1.w + q2 * m2.w + q3 * m3.w;
    float* __restrict__ ad = agg + dst * out + i * 4;
    atomicAdd(ad + 0, vx);
    atomicAdd(ad + 1, vy);
    atomicAdd(ad + 2, vz);
    atomicAdd(ad + 3, vw);
}

// ---------------------------------------------------------------------------
// h = elu(agg * inv_deg + b)   (in place)
// ---------------------------------------------------------------------------
__global__ void finalize_kernel(float* __restrict__ agg, const float* __restrict__ inv_deg,
                                const float* __restrict__ b, int n, int out) {
    int idx = blockIdx.x * blockDim.x + threadIdx.x;     // n*out <= 6.4M
    if (idx >= n * out) return;
    int i = idx / out;
    int col = idx % out;
    agg[idx] = elu1(agg[idx] * inv_deg[i] + b[col]);
}

// ---------------------------------------------------------------------------
// Host-side orchestration
// ---------------------------------------------------------------------------
static inline int gemm_blocks(int M, int Nc, int TN) {
    long waves = (long)(M / 16) * (Nc / (16 * TN));
    return (int)((waves + 7) / 8);             // 256 threads = 8 waves / block
}
static inline int edge_blocks(int total, int c4shift) {
    int epw = 32 >> c4shift;
    long waves = ((long)total + epw - 1) / epw;
    return (int)((waves + 7) / 8);
}

extern "C" void kernel_launch(void* const* d_in, const int* in_sizes, int n_in,
                              void* d_out, int out_size, void* d_ws, size_t ws_size,
                              hipStream_t stream) {
    (void)in_sizes; (void)n_in; (void)out_size; (void)ws_size;
    const int N = NNODE, E = NEDGE;

    const float* x     = (const float*)d_in[0];
    const int*   ei    = (const int*)d_in[1];   // [2,E]: row0=src, row1=dst
    const float* fc0_w = (const float*)d_in[2];
    const float* fc0_b = (const float*)d_in[3];
    const float* W1 = (const float*)d_in[4],  *U1 = (const float*)d_in[5];
    const float* c1 = (const float*)d_in[6],  *b1 = (const float*)d_in[7];
    const float* W2 = (const float*)d_in[8],  *U2 = (const float*)d_in[9];
    const float* c2 = (const float*)d_in[10], *b2 = (const float*)d_in[11];
    const float* W3 = (const float*)d_in[12], *U3 = (const float*)d_in[13];
    const float* c3 = (const float*)d_in[14], *b3 = (const float*)d_in[15];
    const float* fc1_w = (const float*)d_in[16], *fc1_b = (const float*)d_in[17];
    const float* fc2_w = (const float*)d_in[18], *fc2_b = (const float*)d_in[19];

    float* ws    = (float*)d_ws;
    float* invd  = ws;                   // N
    float* P     = ws + (long)N;         // 4N
    float* slab0 = ws + 5L * N;          // 512N : M1/M2/M3, then h4
    float* slab1 = slab0 + 512L * N;     // 128N : h0 -> agg/h per layer
    float* w2pad = slab1 + 128L * N;     // 256*16 padded fc2 weights
    float* b2pad = w2pad + 256 * 16;     // 16 padded fc2 bias

    const int T = 256;

    // degrees
    hipMemsetAsync(invd, 0, (size_t)N * sizeof(float), stream);
    deg_kernel<<<(E + T - 1) / T, T, 0, stream>>>(ei + E, invd, E);
    invdeg_kernel<<<(N + T - 1) / T, T, 0, stream>>>(invd, N);

    // pad fc2 params once (deterministic, re-done every call)
    pad_fc2_kernel<<<(256 * 16 + T - 1) / T, T, 0, stream>>>(fc2_w, fc2_b, w2pad, b2pad);

    // fc0: h0 = elu(x @ fc0_w + fc0_b)          [N,32]->[N,16]
    gemm_wmma<1, true><<<gemm_blocks(N, 16, 1), T, 0, stream>>>(x, fc0_w, fc0_b, slab1, N, 32, 16, 16);

    // ---- FeaSt layer 1: 16 -> 32 ----
    proj_kernel<<<(N + T - 1) / T, T, 0, stream>>>(slab1, U1, P, N, 16);
    gemm_wmma<4, false><<<gemm_blocks(N, 128, 4), T, 0, stream>>>(slab1, W1, nullptr, slab0, N, 16, 128, 128);
    hipMemsetAsync(slab1, 0, (size_t)N * 32 * sizeof(float), stream);
    edge_kernel<<<edge_blocks(E + N, 3), T, 0, stream>>>(ei, E, N, P, c1, slab0, slab1, 32, 3);
    finalize_kernel<<<(N * 32 + T - 1) / T, T, 0, stream>>>(slab1, invd, b1, N, 32);

    // ---- FeaSt layer 2: 32 -> 64 ----
    proj_kernel<<<(N + T - 1) / T, T, 0, stream>>>(slab1, U2, P, N, 32);
    gemm_wmma<4, false><<<gemm_blocks(N, 256, 4), T, 0, stream>>>(slab1, W2, nullptr, slab0, N, 32, 256, 256);
    hipMemsetAsync(slab1, 0, (size_t)N * 64 * sizeof(float), stream);
    edge_kernel<<<edge_blocks(E + N, 4), T, 0, stream>>>(ei, E, N, P, c2, slab0, slab1, 64, 4);
    finalize_kernel<<<(N * 64 + T - 1) / T, T, 0, stream>>>(slab1, invd, b2, N, 64);

    // ---- FeaSt layer 3: 64 -> 128 ----
    proj_kernel<<<(N + T - 1) / T, T, 0, stream>>>(slab1, U3, P, N, 64);
    gemm_wmma<4, false><<<gemm_blocks(N, 512, 4), T, 0, stream>>>(slab1, W3, nullptr, slab0, N, 64, 512, 512);
    hipMemsetAsync(slab1, 0, (size_t)N * 128 * sizeof(float), stream);
    edge_kernel<<<edge_blocks(E + N, 5), T, 0, stream>>>(ei, E, N, P, c3, slab0, slab1, 128, 5);
    finalize_kernel<<<(N * 128 + T - 1) / T, T, 0, stream>>>(slab1, invd, b3, N, 128);

    // fc1: h4 = elu(h3 @ fc1_w + fc1_b)         [N,128]->[N,256]
    gemm_wmma<4, true><<<gemm_blocks(N, 256, 4), T, 0, stream>>>(slab1, fc1_w, fc1_b, slab0, N, 128, 256, 256);

    // fc2: out = h4 @ fc2_w_pad + fc2_b_pad     [N,256]->[N,10] (B padded to 16)
    gemm_wmma<1, false><<<gemm_blocks(N, 16, 1), T, 0, stream>>>(slab0, w2pad, b2pad, (float*)d_out, N, 256, 16, 10);
}